// ECCELoss_80504866996277
// MI455X (gfx1250) — compile-verified
//
#include <hip/hip_runtime.h>
#include <hip/hip_bf16.h>
#include <stdint.h>

// ---------------------------------------------------------------------------
// ECCE (Expected Class Calibration Error) for logits[131072,1000] f32.
// Bandwidth-bound: 524 MB of logits @ 23.3 TB/s => ~22.5us floor.
// Strategy: one wave32 per row; CDNA5 async global->LDS B128 double buffering
// (ASYNCcnt) to prefetch the next row while computing the current one; row
// kept in 32 VGPRs/lane; fast v_exp_f32 path; single f32 atomic per row into
// a 15000-entry (class,bin) diff table; tiny final reduction kernel.
// ---------------------------------------------------------------------------

constexpr int C_DIM  = 1000;          // classes
constexpr int NBINS  = 15;
constexpr int NSEG   = C_DIM * NBINS; // 15000
constexpr int CHUNKS = 8;             // 8 * 128 floats = 1024 >= 1000
constexpr int ROWBUF = CHUNKS * 128;  // floats per LDS row buffer
constexpr int WPB    = 8;             // waves per block (256 threads)
constexpr int BLOCKS = 2048;          // 16384 waves -> 8 rows per wave

#define GLOBAL_AS __attribute__((address_space(1)))
#define LDS_AS    __attribute__((address_space(3)))

typedef int v4i_t __attribute__((ext_vector_type(4)));

#if defined(__HIP_DEVICE_COMPILE__)

#if __has_builtin(__builtin_amdgcn_global_load_async_to_lds_b128)
// Builtin params are int4-vector pointers in global (AS1) / LDS (AS3).
#define ASYNC_CP16(gp, lp)                                                    \
  __builtin_amdgcn_global_load_async_to_lds_b128(                             \
      (GLOBAL_AS v4i_t*)(uintptr_t)(gp),                                      \
      (LDS_AS v4i_t*)(unsigned)(uintptr_t)(lp), 0, 0)
#else
// GV mode: vdst = LDS byte address, v[addr:addr+1] = 64-bit global address.
#define ASYNC_CP16(gp, lp)                                                    \
  do {                                                                        \
    unsigned _l32 = (unsigned)(uintptr_t)(lp);                                \
    asm volatile("global_load_async_to_lds_b128 %0, %1, off"                  \
                 :: "v"(_l32), "v"(gp) : "memory");                           \
  } while (0)
#endif

#if __has_builtin(__builtin_amdgcn_s_wait_asynccnt)
#define WAIT_ASYNC(n)                                                         \
  do { __builtin_amdgcn_s_wait_asynccnt(n);                                   \
       asm volatile("" ::: "memory"); } while (0)
#else
#define WAIT_ASYNC(n) asm volatile("s_wait_asynccnt " #n ::: "memory")
#endif

#else  // host pass: bodies are never codegen'd, keep the parser happy
#define ASYNC_CP16(gp, lp) ((void)(gp), (void)(lp))
#define WAIT_ASYNC(n)      ((void)0)
#endif

// ---------------------------------------------------------------------------

__global__ void ecce_zero_ws(float* __restrict__ p, int n) {
  int i = blockIdx.x * blockDim.x + threadIdx.x;
  if (i < n) p[i] = 0.0f;
}

// Issue 8 async B128 loads covering one 1000-float row (padded to 1024).
// Tail addresses are clamped in-row (valid memory); values masked later.
__device__ __forceinline__ void issue_row_async(const float* __restrict__ rowp,
                                                float* lbuf, int lane) {
#pragma unroll
  for (int i = 0; i < CHUNKS; ++i) {
    int col  = i * 128 + lane * 4;
    int ccol = col > (C_DIM - 4) ? (C_DIM - 4) : col;
    ASYNC_CP16(rowp + ccol, lbuf + col);
  }
}

__device__ __forceinline__ void process_row(const float* lbuf, int lane, int row,
                                            const int* __restrict__ labels,
                                            float* __restrict__ diff) {
  const float NEG_INF = -__builtin_inff();

  // Pull the whole row slice into registers (8 x ds_load_b128 = 32 VGPRs).
  float4 v[CHUNKS];
#pragma unroll
  for (int i = 0; i < CHUNKS; ++i)
    v[i] = *(const float4*)(lbuf + i * 128 + lane * 4);

  // Mask padded tail (only chunk 7 can reach cols >= 1000).
  {
    int base = 7 * 128 + lane * 4;
    if (base + 0 >= C_DIM) v[7].x = NEG_INF;
    if (base + 1 >= C_DIM) v[7].y = NEG_INF;
    if (base + 2 >= C_DIM) v[7].z = NEG_INF;
    if (base + 3 >= C_DIM) v[7].w = NEG_INF;
  }

  // Lane-local max + argmax (ascending cols => strict '>' keeps first index).
  float m  = NEG_INF;
  int   am = 0;
#pragma unroll
  for (int i = 0; i < CHUNKS; ++i) {
    int b = i * 128 + lane * 4;
    if (v[i].x > m) { m = v[i].x; am = b + 0; }
    if (v[i].y > m) { m = v[i].y; am = b + 1; }
    if (v[i].z > m) { m = v[i].z; am = b + 2; }
    if (v[i].w > m) { m = v[i].w; am = b + 3; }
  }
  // wave32 butterfly (max, min-index tie-break) — exact lexicographic reduce.
#pragma unroll
  for (int off = 16; off > 0; off >>= 1) {
    float om = __shfl_xor(m, off, 32);
    int   oa = __shfl_xor(am, off, 32);
    if (om > m || (om == m && oa < am)) { m = om; am = oa; }
  }

  // Sum of exp(v - max); masked slots give exp(-inf) = 0. Fast v_exp_f32.
  float s = 0.0f;
#pragma unroll
  for (int i = 0; i < CHUNKS; ++i) {
    s += __expf(v[i].x - m);
    s += __expf(v[i].y - m);
    s += __expf(v[i].z - m);
    s += __expf(v[i].w - m);
  }
#pragma unroll
  for (int off = 16; off > 0; off >>= 1) s += __shfl_xor(s, off, 32);

  if (lane == 0) {
    float conf = 1.0f / s;                        // max softmax prob
    int   lab  = labels[row];
    int   bin  = (int)ceilf(conf * (float)NBINS) - 1;
    bin = bin < 0 ? 0 : (bin > NBINS - 1 ? NBINS - 1 : bin);
    float accv = (lab == am) ? 1.0f : 0.0f;
    atomicAdd(diff + lab * NBINS + bin, conf - accv);
  }
}

__global__ __launch_bounds__(WPB * 32)
void ecce_rows_kernel(const float* __restrict__ logits,
                      const int* __restrict__ labels,
                      float* __restrict__ diff, int N) {
  __shared__ __align__(16) float lds[WPB * 2 * ROWBUF];  // 64 KB / workgroup

  const int lane   = threadIdx.x & 31;
  const int wave   = threadIdx.x >> 5;
  const int gwave  = blockIdx.x * WPB + wave;
  const int nwaves = gridDim.x * WPB;

  float* buf0 = lds + wave * 2 * ROWBUF;
  float* buf1 = buf0 + ROWBUF;

  int row = gwave;
  if (row < N) issue_row_async(logits + (size_t)row * C_DIM, buf0, lane);

  int k = 0;
  for (; row < N; row += nwaves, ++k) {
    float* cur  = (k & 1) ? buf1 : buf0;
    float* nxt  = (k & 1) ? buf0 : buf1;
    int    nrow = row + nwaves;           // wave-uniform
    if (nrow < N) {
      issue_row_async(logits + (size_t)nrow * C_DIM, nxt, lane);
      WAIT_ASYNC(8);   // the 8 just-issued may be pending; cur's 8 are done
    } else {
      WAIT_ASYNC(0);
    }
    process_row(cur, lane, row, labels, diff);
  }
}

__global__ void ecce_finalize(const float* __restrict__ diff,
                              float* __restrict__ out, float invN) {
  __shared__ float wsum[32];
  int tid = threadIdx.x;
  float s = 0.0f;
  for (int i = tid; i < NSEG; i += 1024) s += fabsf(diff[i]);
#pragma unroll
  for (int off = 16; off > 0; off >>= 1) s += __shfl_xor(s, off, 32);
  if ((tid & 31) == 0) wsum[tid >> 5] = s;
  __syncthreads();
  if (tid < 32) {
    float t = wsum[tid];
#pragma unroll
    for (int off = 16; off > 0; off >>= 1) t += __shfl_xor(t, off, 32);
    if (tid == 0) out[0] = t * invN;
  }
}

// ---------------------------------------------------------------------------

extern "C" void kernel_launch(void* const* d_in, const int* in_sizes, int n_in,
                              void* d_out, int out_size, void* d_ws, size_t ws_size,
                              hipStream_t stream) {
  const float* logits = (const float*)d_in[0];
  const int*   labels = (const int*)d_in[1];
  float*       out    = (float*)d_out;
  float*       diff   = (float*)d_ws;            // NSEG floats = 60 KB

  const int N = in_sizes[1];                     // 131072 rows

  ecce_zero_ws<<<(NSEG + 255) / 256, 256, 0, stream>>>(diff, NSEG);
  ecce_rows_kernel<<<BLOCKS, WPB * 32, 0, stream>>>(logits, labels, diff, N);
  ecce_finalize<<<1, 1024, 0, stream>>>(diff, out, 1.0f / (float)N);
}